// SppPooling_4896262717561
// MI455X (gfx1250) — compile-verified
//
#include <hip/hip_runtime.h>

// Problem constants (from the reference).
#define GRID_N   16
#define K_MAXN   32
#define N_PER    2048
#define D_CH     256
#define R_CELLS  256            // GRID_N * GRID_N
#define XY_STR   35             // 3 + K_MAX ints per node row

#if defined(__gfx1250__) && __has_builtin(__builtin_amdgcn_global_load_async_to_lds_b128)
#define HAVE_ASYNC_LDS 1
#else
#define HAVE_ASYNC_LDS 0
#endif

#if HAVE_ASYNC_LDS
// ---------------------------------------------------------------------------
// CDNA5 path: stage gathered feature rows into LDS with async global->LDS DMA
// (ASYNCcnt-tracked), then reduce from LDS.
//   * 1 block = 1 grid cell, 256 threads (8 wave32).
//   * Staging: lane covers a 16 B chunk (64 chunks per 1 KiB row); the block
//     moves 4 rows per pass, so all <=32 rows are in flight before the wait.
//   * Reduce: thread t = channel t; consecutive threads hit consecutive LDS
//     banks (stride-1 within a row) -> conflict-free.
// ---------------------------------------------------------------------------
typedef int v4i __attribute__((ext_vector_type(4)));
typedef v4i __attribute__((address_space(1))) * gptr_v4i;  // global v4i*
typedef v4i __attribute__((address_space(3))) * lptr_v4i;  // LDS v4i*

__global__ void spp_pool_async(const float* __restrict__ features,
                               const int*   __restrict__ xy,
                               float*       __restrict__ out) {
    const int b = blockIdx.x >> 8;
    const int r = blockIdx.x & 255;
    const int t = threadIdx.x;

    __shared__ int   s_idx[K_MAXN];
    __shared__ int   s_meta[3];
    __shared__ float s_rows[K_MAXN * D_CH];   // 32 KiB staging buffer

    const long long xyBase = ((long long)b * N_PER + r) * XY_STR;
    if (t < 3 + K_MAXN) {
        int v = xy[xyBase + t];
        if (t < 3) s_meta[t] = v; else s_idx[t - 3] = v;
    }
    __syncthreads();

    const int row = s_meta[0];
    const int col = s_meta[1];
    int c = s_meta[2];
    c = c < 0 ? 0 : (c > K_MAXN ? K_MAXN : c);

    const int chunk = t & 63;    // 16-byte chunk within a row
    const int slot  = t >> 6;    // row within a 4-row pass (wave-uniform)
    const float* fbase = features + (long long)b * N_PER * D_CH;

    for (int base = 0; base < c; base += 4) {
        const int rr = base + slot;
        if (rr < c) {            // wave-uniform branch -> EXEC all-ones inside
            float* src = const_cast<float*>(fbase + (long long)s_idx[rr] * D_CH + chunk * 4);
            float* dst = &s_rows[rr * D_CH + chunk * 4];
            __builtin_amdgcn_global_load_async_to_lds_b128(
                (gptr_v4i)src, (lptr_v4i)dst, /*offset=*/0, /*cpol=*/0);
        }
    }
#if __has_builtin(__builtin_amdgcn_s_wait_asynccnt)
    __builtin_amdgcn_s_wait_asynccnt(0);
#else
    asm volatile("s_wait_asynccnt 0" ::: "memory");
#endif
    __syncthreads();             // make every wave's staged rows visible

    if (row >= 0) {
        float acc = 0.f;
        #pragma unroll 4
        for (int k = 0; k < c; ++k)
            acc += s_rows[k * D_CH + t];
        const float denom = (float)(c > 0 ? c : 1);
        out[((long long)b * D_CH + t) * R_CELLS + (row * GRID_N + col)] = acc / denom;
    }
}

#else  // !HAVE_ASYNC_LDS -------------------------------------------------------
// Fallback: direct gather-accumulate. 1 block = 1 cell, 256 threads = channels.
__global__ void spp_pool_direct(const float* __restrict__ features,
                                const int*   __restrict__ xy,
                                float*       __restrict__ out) {
    const int b = blockIdx.x >> 8;
    const int r = blockIdx.x & 255;
    const int t = threadIdx.x;

    __shared__ int s_idx[K_MAXN];
    __shared__ int s_meta[3];

    const long long xyBase = ((long long)b * N_PER + r) * XY_STR;
    if (t < 3 + K_MAXN) {
        int v = xy[xyBase + t];
        if (t < 3) s_meta[t] = v; else s_idx[t - 3] = v;
    }
    __syncthreads();

    const int row = s_meta[0];
    const int col = s_meta[1];
    int c = s_meta[2];
    c = c < 0 ? 0 : (c > K_MAXN ? K_MAXN : c);
    if (row < 0) return;

    const float* fb = features + (long long)b * N_PER * D_CH + t;
    float acc = 0.f;
    int k = 0;
    for (; k + 4 <= c; k += 4) {
        float a0 = fb[(long long)s_idx[k + 0] * D_CH];
        float a1 = fb[(long long)s_idx[k + 1] * D_CH];
        float a2 = fb[(long long)s_idx[k + 2] * D_CH];
        float a3 = fb[(long long)s_idx[k + 3] * D_CH];
        acc += (a0 + a1) + (a2 + a3);
    }
    for (; k < c; ++k)
        acc += fb[(long long)s_idx[k] * D_CH];

    const float denom = (float)(c > 0 ? c : 1);
    out[((long long)b * D_CH + t) * R_CELLS + (row * GRID_N + col)] = acc / denom;
}
#endif

// ---------------------------------------------------------------------------
extern "C" void kernel_launch(void* const* d_in, const int* in_sizes, int n_in,
                              void* d_out, int out_size, void* d_ws, size_t ws_size,
                              hipStream_t stream) {
    const float* features = (const float*)d_in[0];
    const int*   xy       = (const int*)d_in[1];
    // d_in[2] (nodes_per_graph) is constant N_PER in this problem; unused.
    float* out = (float*)d_out;

    const int B = (n_in > 2 && in_sizes) ? in_sizes[2] : 64;  // batch count
    dim3 grid((unsigned)(B * R_CELLS));
    dim3 block(256);

#if HAVE_ASYNC_LDS
    spp_pool_async<<<grid, block, 0, stream>>>(features, xy, out);
#else
    spp_pool_direct<<<grid, block, 0, stream>>>(features, xy, out);
#endif
}